// FlyVisHodgkinHuxleyODE_34677565948816
// MI455X (gfx1250) — compile-verified
//
#include <hip/hip_runtime.h>
#include <stdint.h>

// ---------------------------------------------------------------------------
// FlyVis Hodgkin-Huxley ODE step:  dv = (-I_Na - I_K - I_L + I_syn + I_ext)/C
// Edge part: I_syn[dst] += W * sigmoid((v[src]-v_half[src])/slope[src])
// Sigmoid is hoisted to per-node precompute (100K exp instead of 6.4M).
// Edge streaming (src,dst,W) goes through the CDNA5 async global->LDS DMA
// path (non-temporal), double buffered with s_wait_asynccnt.
// ---------------------------------------------------------------------------

#define TPB   256          // threads per block (8 wave32)
#define EPT   4            // edges per thread per tile (128-bit chunks)
#define TILE  (TPB * EPT)  // 1024 edges per tile

#define ASYNC_CPOL_NT 1    // CPol TH[2:0]=NT: single-use stream, don't churn L2

typedef int v4i __attribute__((ext_vector_type(4)));
typedef __attribute__((address_space(1))) v4i gv4i;   // global int4
typedef __attribute__((address_space(3))) v4i lv4i;   // LDS int4

__device__ __forceinline__ void async_copy_b128(const void* gsrc, void* ldst) {
#if __has_builtin(__builtin_amdgcn_global_load_async_to_lds_b128)
  __builtin_amdgcn_global_load_async_to_lds_b128(
      (gv4i*)(uintptr_t)gsrc,
      (lv4i*)(uint32_t)(uintptr_t)ldst,   // low 32 bits of flat LDS addr = LDS offset
      /*offset=*/0, /*cpol=*/ASYNC_CPOL_NT);
#else
  unsigned lofs = (uint32_t)(uintptr_t)ldst;
  asm volatile("global_load_async_to_lds_b128 %0, %1, off th:TH_LOAD_NT"
               :: "v"(lofs), "v"(gsrc) : "memory");
#endif
}

__device__ __forceinline__ void wait_asynccnt_3() {
#if __has_builtin(__builtin_amdgcn_s_wait_asynccnt)
  __builtin_amdgcn_s_wait_asynccnt(3);
#else
  asm volatile("s_wait_asynccnt 3" ::: "memory");
#endif
}

__device__ __forceinline__ void wait_asynccnt_0() {
#if __has_builtin(__builtin_amdgcn_s_wait_asynccnt)
  __builtin_amdgcn_s_wait_asynccnt(0);
#else
  asm volatile("s_wait_asynccnt 0" ::: "memory");
#endif
}

__device__ __forceinline__ void atomic_fadd_agent(float* p, float v) {
  // Relaxed, agent scope, result unused -> no-return global_atomic_add_f32
  __hip_atomic_fetch_add(p, v, __ATOMIC_RELAXED, __HIP_MEMORY_SCOPE_AGENT);
}

// ---------------------------------------------------------------------------
// Edge scatter (placed FIRST so the disasm snippet shows it). Persistent
// blocks; each tile's src/dst/W is pulled into LDS with async DMA (double
// buffered). Each thread owns its 4 edges (writes and reads its own 16B LDS
// slots -> no workgroup barrier needed; the async->consume hazard is covered
// by s_wait_asynccnt).
// ---------------------------------------------------------------------------
__global__ __launch_bounds__(TPB) void edge_kernel(
    const int* __restrict__ src, const int* __restrict__ dst,
    const float* __restrict__ W, const float* __restrict__ act,
    float* __restrict__ accum, long long E, int ntiles) {
  __shared__ int4   s_src[2][TPB];
  __shared__ int4   s_dst[2][TPB];
  __shared__ float4 s_w[2][TPB];
  const int tid = threadIdx.x;

  int t0 = blockIdx.x;
  if (t0 < ntiles) {
    long long base = (long long)t0 * TILE + tid * EPT;
    async_copy_b128(src + base, &s_src[0][tid]);
    async_copy_b128(dst + base, &s_dst[0][tid]);
    async_copy_b128(W + base,   &s_w[0][tid]);
  }

  int it = 0;
  for (int t = t0; t < ntiles; t += (int)gridDim.x, ++it) {
    const int b = it & 1;
    const int tn = t + (int)gridDim.x;
    if (tn < ntiles) {
      // prefetch next tile into the other buffer, then wait for current tile
      long long base = (long long)tn * TILE + tid * EPT;
      async_copy_b128(src + base, &s_src[b ^ 1][tid]);
      async_copy_b128(dst + base, &s_dst[b ^ 1][tid]);
      async_copy_b128(W + base,   &s_w[b ^ 1][tid]);
      wait_asynccnt_3();   // first 3 (current tile) retired; 3 in flight
    } else {
      wait_asynccnt_0();
    }
    int4   s4 = s_src[b][tid];
    int4   d4 = s_dst[b][tid];
    float4 w4 = s_w[b][tid];

    float m0 = w4.x * act[s4.x];
    float m1 = w4.y * act[s4.y];
    float m2 = w4.z * act[s4.z];
    float m3 = w4.w * act[s4.w];
    atomic_fadd_agent(&accum[d4.x], m0);
    atomic_fadd_agent(&accum[d4.y], m1);
    atomic_fadd_agent(&accum[d4.z], m2);
    atomic_fadd_agent(&accum[d4.w], m3);
  }

  // tail edges (E % TILE) via plain loads
  long long tail0 = (long long)ntiles * TILE;
  long long stride = (long long)gridDim.x * blockDim.x;
  for (long long e = tail0 + (long long)blockIdx.x * blockDim.x + tid; e < E; e += stride) {
    int s = src[e];
    int d = dst[e];
    float msg = W[e] * act[s];
    atomic_fadd_agent(&accum[d], msg);
  }
}

// ---------------------------------------------------------------------------
// Per-node precompute.
//   act[i]   = sigmoid((v - v_half)/slope)
//   accum[i] = I_ext - I_Na - I_K - I_L        (atomics add I_syn onto this)
// ---------------------------------------------------------------------------
__global__ __launch_bounds__(TPB) void node_prep(
    const float* __restrict__ v, const float* __restrict__ stim,
    const float* __restrict__ m, const float* __restrict__ h,
    const float* __restrict__ n, const float* __restrict__ vhalf,
    const float* __restrict__ slope, const float* __restrict__ gNa,
    const float* __restrict__ gK, const float* __restrict__ gL,
    const float* __restrict__ ENa, const float* __restrict__ EK,
    const float* __restrict__ EL, const float* __restrict__ Ibias,
    const float* __restrict__ sscale,
    float* __restrict__ act, float* __restrict__ accum, int N) {
  int i = blockIdx.x * blockDim.x + threadIdx.x;
  if (i >= N) return;
  float vi = v[i];
  float mi = m[i];
  float m3 = mi * mi * mi;
  float ni = n[i];
  float n2 = ni * ni;
  float n4 = n2 * n2;
  float INa = gNa[i] * m3 * h[i] * (vi - ENa[i]);
  float IK  = gK[i] * n4 * (vi - EK[i]);
  float IL  = gL[i] * (vi - EL[i]);
  float Iext = Ibias[i] + sscale[i] * stim[i];
  accum[i] = Iext - INa - IK - IL;
  float x = (vi - vhalf[i]) / slope[i];
  act[i] = 1.0f / (1.0f + expf(-x));
}

// ---------------------------------------------------------------------------
// dv = accum / C
// ---------------------------------------------------------------------------
__global__ __launch_bounds__(TPB) void finalize(
    const float* __restrict__ accum, const float* __restrict__ C,
    float* __restrict__ out, int N) {
  int i = blockIdx.x * blockDim.x + threadIdx.x;
  if (i >= N) return;
  out[i] = accum[i] / C[i];
}

// ---------------------------------------------------------------------------
// Launch
// ---------------------------------------------------------------------------
extern "C" void kernel_launch(void* const* d_in, const int* in_sizes, int n_in,
                              void* d_out, int out_size, void* d_ws, size_t ws_size,
                              hipStream_t stream) {
  const float* voltage  = (const float*)d_in[0];
  const float* stimulus = (const float*)d_in[1];
  const float* hh_m     = (const float*)d_in[2];
  const float* hh_h     = (const float*)d_in[3];
  const float* hh_n     = (const float*)d_in[4];
  const float* W        = (const float*)d_in[5];
  const float* vhalf    = (const float*)d_in[6];
  const float* slope    = (const float*)d_in[7];
  const float* gNa      = (const float*)d_in[8];
  const float* gK       = (const float*)d_in[9];
  const float* gL       = (const float*)d_in[10];
  const float* ENa      = (const float*)d_in[11];
  const float* EK       = (const float*)d_in[12];
  const float* EL       = (const float*)d_in[13];
  const float* C        = (const float*)d_in[14];
  const float* Ibias    = (const float*)d_in[15];
  const float* sscale   = (const float*)d_in[16];
  const int*   eidx     = (const int*)d_in[17];

  const int       N = in_sizes[0];
  const long long E = (long long)in_sizes[5];
  const int*      src = eidx;        // edge_index[0]
  const int*      dst = eidx + E;    // edge_index[1]

  float* act   = (float*)d_ws;       // N floats
  float* accum = act + N;            // N floats

  const int nblocks_node = (N + TPB - 1) / TPB;
  node_prep<<<nblocks_node, TPB, 0, stream>>>(
      voltage, stimulus, hh_m, hh_h, hh_n, vhalf, slope, gNa, gK, gL,
      ENa, EK, EL, Ibias, sscale, act, accum, N);

  const int ntiles = (int)(E / TILE);
  int nblocks_edge = 2048;
  if (ntiles > 0 && ntiles < nblocks_edge) nblocks_edge = ntiles;
  if (nblocks_edge < 1) nblocks_edge = 1;
  edge_kernel<<<nblocks_edge, TPB, 0, stream>>>(
      src, dst, W, act, accum, E, ntiles);

  finalize<<<nblocks_node, TPB, 0, stream>>>(accum, C, (float*)d_out, N);
}